// TopKGateParallel_62354335203867
// MI455X (gfx1250) — compile-verified
//
#include <hip/hip_runtime.h>
#include <math.h>

typedef __attribute__((ext_vector_type(2))) float v2f;
typedef __attribute__((ext_vector_type(8))) float v8f;
typedef __attribute__((ext_vector_type(4))) int   v4i;

#define N_TOK   32768
#define D_DIM   1024
#define E_DIM   64
#define TOPK    8
#define BLK_ROWS 32
#define XS_STR  68   // padded LDS row stride (floats) for x / gate_w / noise chunks
#define LG_STR  69   // padded LDS row stride for logits / sparse weights
#define NCHUNK  (D_DIM / 64)

#if __has_builtin(__builtin_amdgcn_global_load_async_to_lds_b128) && \
    __has_builtin(__builtin_amdgcn_s_wait_asynccnt)
#define USE_ASYNC 1
#endif

#ifdef USE_ASYNC
typedef __attribute__((address_space(1))) v4i* gptr_v4i;   // global (AS1) int4*
typedef __attribute__((address_space(3))) v4i* lptr_v4i;   // LDS (AS3) int4*
__device__ __forceinline__ gptr_v4i gas(const void* p) {
    return (gptr_v4i)(unsigned long long)p;    // global ptr bits identical
}
__device__ __forceinline__ lptr_v4i las(const void* p) {
    // generic LDS pointer: low 32 bits are the LDS offset (flat aperture mapping)
    return (lptr_v4i)(unsigned)(unsigned long long)p;
}
#endif

__global__ void moe_init_kernel(float* __restrict__ gsum) {
    gsum[threadIdx.x] = 0.0f;
}

// Stage one 64-wide K chunk of x [32x64] and gate_w [64x64] into LDS.
__device__ __forceinline__ void stage_chunk(const float* __restrict__ x,
                                            const float* __restrict__ gw,
                                            int n0, int dc, int t,
                                            float* xbuf, float* wbuf)
{
#ifdef USE_ASYNC
    #pragma unroll
    for (int i = 0; i < 2; ++i) {
        int idx = t + i * 256;
        int r = idx >> 4, c = (idx & 15) << 2;
        __builtin_amdgcn_global_load_async_to_lds_b128(
            gas(x + (size_t)(n0 + r) * D_DIM + dc + c),
            las(xbuf + r * XS_STR + c), 0, 0);
    }
    #pragma unroll
    for (int i = 0; i < 4; ++i) {
        int idx = t + i * 256;
        int r = idx >> 4, c = (idx & 15) << 2;
        __builtin_amdgcn_global_load_async_to_lds_b128(
            gas(gw + (size_t)r * D_DIM + dc + c),
            las(wbuf + r * XS_STR + c), 0, 0);
    }
#else
    #pragma unroll
    for (int i = 0; i < 2; ++i) {
        int idx = t + i * 256;
        int r = idx >> 4, c = (idx & 15) << 2;
        *reinterpret_cast<float4*>(xbuf + r * XS_STR + c) =
            *reinterpret_cast<const float4*>(x + (size_t)(n0 + r) * D_DIM + dc + c);
    }
    #pragma unroll
    for (int i = 0; i < 4; ++i) {
        int idx = t + i * 256;
        int r = idx >> 4, c = (idx & 15) << 2;
        *reinterpret_cast<float4*>(wbuf + r * XS_STR + c) =
            *reinterpret_cast<const float4*>(gw + (size_t)r * D_DIM + dc + c);
    }
#endif
}

__global__ __launch_bounds__(256) void moe_gate_kernel(
    const float* __restrict__ x,      // [N, D]
    const float* __restrict__ gw,     // [E, D]
    const float* __restrict__ nwg,    // [E]
    const float* __restrict__ noise,  // [N, E]
    float* __restrict__ out_w,        // [N, E] sparse gated weights
    float* __restrict__ out_ids,      // [N, K] ids as float
    float* __restrict__ gsum)         // [E] global softmax-mean accumulator
{
    __shared__ float xs [2][BLK_ROWS * XS_STR];  // x chunk double buffer
    __shared__ float wsm[2][E_DIM    * XS_STR];  // gate_w double buffer (buf0 reused for noise)
    __shared__ float lg [BLK_ROWS * LG_STR];     // logits -> sparse weights
    __shared__ float ps [E_DIM];                 // per-block softmax partial sums
    __shared__ float nw [E_DIM];                 // noise_weight
    __shared__ float idb[BLK_ROWS * TOPK];       // staged ids

    const int t    = threadIdx.x;
    const int n0   = blockIdx.x * BLK_ROWS;
    const int wid  = t >> 5;
    const int lane = t & 31;
    const int lm   = lane & 15;       // M (A-rows) / N (B-cols) within fragment
    const int lh   = lane >> 4;       // lane half -> K offset 2*lh
    const int lr0  = (wid & 1) * 16;  // local row tile
    const int e0   = (wid >> 1) * 16; // expert tile

    if (t < E_DIM) { nw[t] = nwg[t]; ps[t] = 0.0f; }

    v8f acc = {};

    // Prologue: kick off chunk 0.
    stage_chunk(x, gw, n0, 0, t, xs[0], wsm[0]);

    for (int it = 0; it < NCHUNK; ++it) {
        const int buf = it & 1;
#ifdef USE_ASYNC
        __builtin_amdgcn_s_wait_asynccnt(0);   // own async writes for chunk `it` done
#endif
        __syncthreads();                        // chunk `it` visible; buf^1 free to refill
        if (it + 1 < NCHUNK)
            stage_chunk(x, gw, n0, (it + 1) * 64, t, xs[buf ^ 1], wsm[buf ^ 1]);

        const float* xb = xs[buf];
        const float* wb = wsm[buf];
        // ---- 16 x V_WMMA_F32_16X16X4_F32 over this 64-wide K chunk ----
        #pragma unroll
        for (int kb = 0; kb < 64; kb += 4) {
            v2f a = *reinterpret_cast<const v2f*>(xb + (lr0 + lm) * XS_STR + kb + 2 * lh);
            v2f b = *reinterpret_cast<const v2f*>(wb + (e0  + lm) * XS_STR + kb + 2 * lh);
            acc = __builtin_amdgcn_wmma_f32_16x16x4_f32(
                false, a, false, b, (short)0, acc, false, false);
        }
    }

    // ---- stage noise chunk [32 x 64] into retired wsm[0] ----
#ifdef USE_ASYNC
    #pragma unroll
    for (int i = 0; i < 2; ++i) {
        int idx = t + i * 256;
        int r = idx >> 4, c = (idx & 15) << 2;
        __builtin_amdgcn_global_load_async_to_lds_b128(
            gas(noise + (size_t)(n0 + r) * E_DIM + c),
            las(&wsm[0][0] + r * XS_STR + c), 0, 0);
    }
#else
    #pragma unroll
    for (int i = 0; i < 2; ++i) {
        int idx = t + i * 256;
        int r = idx >> 4, c = (idx & 15) << 2;
        *reinterpret_cast<float4*>(&wsm[0][0] + r * XS_STR + c) =
            *reinterpret_cast<const float4*>(noise + (size_t)(n0 + r) * E_DIM + c);
    }
#endif

    // ---- spill logits tile to LDS (C/D layout: VGPR v -> M = v + 8*lh, N = lm) ----
    #pragma unroll
    for (int v = 0; v < 8; ++v)
        lg[(lr0 + v + 8 * lh) * LG_STR + e0 + lm] = acc[v];

#ifdef USE_ASYNC
    __builtin_amdgcn_s_wait_asynccnt(0);
#endif
    __syncthreads();

    // ---- per-row epilogue: dense softmax mean, noisy top-8, sparse softmax ----
    if (t < BLK_ROWS) {
        float* row        = lg + t * LG_STR;
        const float* nrow = &wsm[0][0] + t * XS_STR;

        float mx = -INFINITY;
        for (int e = 0; e < E_DIM; ++e) mx = fmaxf(mx, row[e]);
        float sum = 0.0f;
        for (int e = 0; e < E_DIM; ++e) sum += __expf(row[e] - mx);
        const float inv = 1.0f / sum;
        // rotated index -> conflict-free ds_add_f32 accumulation
        for (int i = 0; i < E_DIM; ++i) {
            int e = (i + 2 * t) & (E_DIM - 1);
            atomicAdd(&ps[e], __expf(row[e] - mx) * inv);
        }

        // noisy top-8 (stable insertion: strict '>' keeps lowest index on ties)
        float tv[TOPK]; int ti[TOPK];
        #pragma unroll
        for (int j = 0; j < TOPK; ++j) { tv[j] = -INFINITY; ti[j] = 0; }
        for (int e = 0; e < E_DIM; ++e) {
            float v = row[e] + nrow[e] * nw[e];
            if (v > tv[TOPK - 1]) {
                int j = TOPK - 1;
                while (j > 0 && v > tv[j - 1]) {
                    tv[j] = tv[j - 1]; ti[j] = ti[j - 1]; --j;
                }
                tv[j] = v; ti[j] = e;
            }
        }
        // sparse softmax over the 8 selected noisy logits
        float w[TOPK]; float s2 = 0.0f;
        #pragma unroll
        for (int j = 0; j < TOPK; ++j) { w[j] = __expf(tv[j] - tv[0]); s2 += w[j]; }
        const float is2 = 1.0f / s2;
        for (int e = 0; e < E_DIM; ++e) row[e] = 0.0f;
        #pragma unroll
        for (int j = 0; j < TOPK; ++j) {
            row[ti[j]]        = w[j] * is2;
            idb[t * TOPK + j] = (float)ti[j];
        }
    }
    __syncthreads();

    if (t < E_DIM) atomicAdd(&gsum[t], ps[t]);

    // ---- coalesced writeout: sparse weights [32 x 64] as float4, ids flat ----
    #pragma unroll
    for (int k = 0; k < 2; ++k) {
        int f = t + k * 256;               // float4 slot in [0, 512)
        int r = f >> 4, c = (f & 15) << 2;
        float4 v;
        v.x = lg[r * LG_STR + c + 0];
        v.y = lg[r * LG_STR + c + 1];
        v.z = lg[r * LG_STR + c + 2];
        v.w = lg[r * LG_STR + c + 3];
        *reinterpret_cast<float4*>(out_w + (size_t)(n0 + r) * E_DIM + c) = v;
    }
    out_ids[(size_t)n0 * TOPK + t] = idb[t];
}

__global__ void moe_loss_kernel(const float* __restrict__ gsum,
                                float* __restrict__ out_loss) {
    __shared__ float red[E_DIM];
    const int t = threadIdx.x;
    float d = gsum[t] * (1.0f / (float)N_TOK) - (1.0f / (float)E_DIM);
    red[t] = d * d;
    __syncthreads();
    for (int s = E_DIM / 2; s > 0; s >>= 1) {
        if (t < s) red[t] += red[t + s];
        __syncthreads();
    }
    if (t == 0) *out_loss = (red[0] / (float)E_DIM) * 0.01f;
}

extern "C" void kernel_launch(void* const* d_in, const int* in_sizes, int n_in,
                              void* d_out, int out_size, void* d_ws, size_t ws_size,
                              hipStream_t stream) {
    const float* x     = (const float*)d_in[0];
    const float* gw    = (const float*)d_in[1];
    const float* nwg   = (const float*)d_in[2];
    const float* noise = (const float*)d_in[3];

    float* out      = (float*)d_out;
    float* out_w    = out;                                                  // N*E
    float* out_ids  = out + (size_t)N_TOK * E_DIM;                          // N*K
    float* out_loss = out + (size_t)N_TOK * E_DIM + (size_t)N_TOK * TOPK;   // 1
    float* gsum     = (float*)d_ws;                                         // E floats

    moe_init_kernel<<<dim3(1), dim3(E_DIM), 0, stream>>>(gsum);
    moe_gate_kernel<<<dim3(N_TOK / BLK_ROWS), dim3(256), 0, stream>>>(
        x, gw, nwg, noise, out_w, out_ids, gsum);
    moe_loss_kernel<<<dim3(1), dim3(E_DIM), 0, stream>>>(gsum, out_loss);
}